// TrackAwareEvidenceChain_57758720197311
// MI455X (gfx1250) — compile-verified
//
#include <hip/hip_runtime.h>
#include <hip/hip_bf16.h>
#include <stdint.h>

// Problem constants (from reference setup_inputs)
#define B_    8
#define L_    4096
#define D_    1024
#define T_    64                 // rows per scan chunk
#define NC_   (L_ / T_)          // 64 chunks per batch sequence
#define NROWS (B_ * L_)          // 32768 rows
#define TPB   256                // 8 wave32 waves
#define WAVES (TPB / 32)

#define RT_   8                  // rows per TDM tile in K2a
#define NT_   (T_ / RT_)         // 8 tiles per chunk

#define LN_EPS   1e-5f
#define NORM_EPS 1e-4f

typedef unsigned int u32x4 __attribute__((ext_vector_type(4)));
typedef int          i32x8 __attribute__((ext_vector_type(8)));
typedef int          i32x4 __attribute__((ext_vector_type(4)));

__device__ __forceinline__ float wave_reduce_add(float v) {
#pragma unroll
    for (int off = 16; off > 0; off >>= 1)
        v += __shfl_xor(v, off, 32);
    return v;
}

// lane-wide async global->LDS 16B load (GV mode), tracked with ASYNCcnt
__device__ __forceinline__ void async_b128(const float* gptr, const float* lptr) {
    const uint32_t lds = (uint32_t)(uintptr_t)lptr;     // low 32 bits = LDS byte addr
    const uint64_t ga  = (uint64_t)(uintptr_t)gptr;
    asm volatile("global_load_async_to_lds_b128 %0, %1, off"
                 :: "v"(lds), "v"(ga) : "memory");
}

// TDM: DMA a rows x 1024 f32 tile (row-major, stride 1024) into LDS.
// D# built per CDNA5 ISA §8 (groups 2/3 zero => 2-D tensor).
// amdgpu-toolchain (clang-23) 6-arg builtin form.
__device__ __forceinline__ void tdm_load_tile(const float* gptr, const float* lptr,
                                              int rows) {
    const uint64_t ga  = (uint64_t)(uintptr_t)gptr;
    const uint32_t lds = (uint32_t)(uintptr_t)lptr;
    u32x4 g0;
    g0[0] = 1u;                                           // count=1, user descriptor
    g0[1] = lds;                                          // lds_addr
    g0[2] = (uint32_t)ga;                                 // global_addr[31:0]
    g0[3] = (uint32_t)((ga >> 32) & 0x1FFFFFFu)           // global_addr[56:32]
          | 0x80000000u;                                  // type=2 ("image")
    i32x8 g1;
    g1[0] = 0x00020000;                                   // data_size=2 (4B), no flags
    g1[1] = (int)(1024u << 16);                           // tensor_dim0 = 1024
    g1[2] = (int)((uint32_t)rows << 16);                  // tensor_dim1 = rows
    g1[3] = (int)(1024u << 16);                           // tile_dim0 = 1024
    g1[4] = rows;                                         // tile_dim1 = rows
    g1[5] = 1024;                                         // tensor_dim0_stride = 1024
    g1[6] = 0;
    g1[7] = 0;
    i32x4 z4; z4[0] = 0; z4[1] = 0; z4[2] = 0; z4[3] = 0;
    i32x8 z8;
#pragma unroll
    for (int i = 0; i < 8; ++i) z8[i] = 0;
    __builtin_amdgcn_tensor_load_to_lds(g0, g1, z4, z4, z8, 0);
}

// ---------------------------------------------------------------------------
// K1: per-row LayerNorm stats + transition gates + within-chunk cumulative
//     gate products. Rows stream through LDS via async loads (ASYNCcnt),
//     double buffered; the previous row lives in registers.
// ---------------------------------------------------------------------------
__global__ __launch_bounds__(TPB)
void k1_gates(const float* __restrict__ x,
              const float* __restrict__ bias_p,
              const float* __restrict__ scale_p,
              float* __restrict__ gate,
              float* __restrict__ rowmu,
              float* __restrict__ rowrstd,
              float* __restrict__ avec,
              float* __restrict__ aend)
{
    const int blk  = blockIdx.x;
    const int b    = blk / NC_;
    const int c    = blk % NC_;
    const int tid  = threadIdx.x;
    const int wave = tid >> 5;
    const int lane = tid & 31;
    const long base = (long)b * L_ + (long)c * T_;

    __shared__ float sh[2][D_];                      // two 4KB row buffers
    __shared__ float s1s[WAVES], s2s[WAVES], s3s[WAVES];

    // load i: i==0 -> prev row, i>=1 -> row base+i-1; buffer i&1
    const long rprev = (c > 0) ? (base - 1) : base;  // self-dup at sequence start
    async_b128(x + rprev * D_ + tid * 4, &sh[0][tid * 4]);
    async_b128(x + base  * D_ + tid * 4, &sh[1][tid * 4]);
    asm volatile("s_wait_asynccnt 1" ::: "memory");  // prev row landed

    float4 xp = *(const float4*)(&sh[0][tid * 4]);

    float mu_p = 0.f, r_p = 0.f, n_p = 0.f;          // valid in thread 0 only
    float tb = 0.f, sps = 0.f, A = 1.0f;             // valid in thread 0 only
    if (tid == 0) {
        tb = *bias_p;
        const float sc = *scale_p;
        sps = (sc > 20.0f) ? sc : log1pf(expf(sc));  // softplus(transition_scale)
    }

    // stats of the previous row
    {
        float s1 = xp.x + xp.y + xp.z + xp.w;
        float s2 = xp.x*xp.x + xp.y*xp.y + xp.z*xp.z + xp.w*xp.w;
        s1 = wave_reduce_add(s1);
        s2 = wave_reduce_add(s2);
        if (lane == 0) { s1s[wave] = s1; s2s[wave] = s2; }
        __syncthreads();
        if (tid == 0) {
            float S1 = 0.f, S2 = 0.f;
#pragma unroll
            for (int w = 0; w < WAVES; ++w) { S1 += s1s[w]; S2 += s2s[w]; }
            mu_p = S1 * (1.0f / D_);
            float var = fmaxf(S2 * (1.0f / D_) - mu_p * mu_p, 0.0f);
            r_p = rsqrtf(var + LN_EPS);
            n_p = sqrtf((float)D_ * var) * r_p;      // ||h_prev||
        }
        __syncthreads();
    }

    for (int k = 0; k < T_; ++k) {
        // prefetch row base+k+1 (load index k+2) into buf[k&1]; that buffer's
        // previous contents were consumed into registers in iteration k-1
        // (before its reduction barriers), so the overwrite is safe.
        if (k + 1 < T_) {
            async_b128(x + (base + k + 1) * D_ + tid * 4, &sh[k & 1][tid * 4]);
            asm volatile("s_wait_asynccnt 1" ::: "memory");
        } else {
            asm volatile("s_wait_asynccnt 0" ::: "memory");
        }

        const long r = base + k;
        const float4 v = *(const float4*)(&sh[(k + 1) & 1][tid * 4]);
        float s1 = v.x + v.y + v.z + v.w;
        float s2 = v.x*v.x + v.y*v.y + v.z*v.z + v.w*v.w;
        float s3 = v.x*xp.x + v.y*xp.y + v.z*xp.z + v.w*xp.w;
        s1 = wave_reduce_add(s1);
        s2 = wave_reduce_add(s2);
        s3 = wave_reduce_add(s3);
        if (lane == 0) { s1s[wave] = s1; s2s[wave] = s2; s3s[wave] = s3; }
        __syncthreads();
        if (tid == 0) {
            float S1 = 0.f, S2 = 0.f, S3 = 0.f;
#pragma unroll
            for (int w = 0; w < WAVES; ++w) { S1 += s1s[w]; S2 += s2s[w]; S3 += s3s[w]; }
            const float mu  = S1 * (1.0f / D_);
            const float var = fmaxf(S2 * (1.0f / D_) - mu * mu, 0.0f);
            const float rs  = rsqrtf(var + LN_EPS);
            const float nt  = sqrtf((float)D_ * var) * rs;               // ||h_t||
            const float dot = rs * r_p * (S3 - (float)D_ * mu * mu_p);   // h_t . h_prev
            const float qnt = fmaxf(nt,  NORM_EPS);
            const float qnp = fmaxf(n_p, NORM_EPS);
            const float qq  = dot / (qnt * qnp);                         // q_t . q_prev
            const float sqt = (nt * nt) / (qnt * qnt);
            const float sqp = (n_p * n_p) / (qnp * qnp);
            const float cosd = 1.0f - qq;
            const float msq  = (sqt + sqp - 2.0f * qq) * (1.0f / D_);
            const float mag  = (msq > 0.0f) ? sqrtf(msq) : 0.0f;
            const float score = cosd + 0.5f * mag;
            const float z = tb - sps * score;
            const float stay = 1.0f / (1.0f + expf(-z));
            const float a = (c == 0 && k == 0) ? 0.0f : stay;
            A *= a;
            gate[r]    = a;
            avec[r]    = A;
            rowmu[r]   = mu;
            rowrstd[r] = rs;
            mu_p = mu; r_p = rs; n_p = nt;
            if (k == T_ - 1) aend[b * NC_ + c] = A;
        }
        __syncthreads();
        xp = v;
    }
}

// ---------------------------------------------------------------------------
// K2a: local (within-chunk) scan, zero carry-in. 8-row x 4KB tiles are DMAed
//      into LDS by the Tensor Data Mover, double buffered, synchronized with
//      TENSORcnt + workgroup barriers. Local scan results go to d_out
//      (scratch); chunk-end state to lend.
// ---------------------------------------------------------------------------
__global__ __launch_bounds__(TPB)
void k2a_localscan(const float* __restrict__ x,
                   const float* __restrict__ rowmu,
                   const float* __restrict__ rowrstd,
                   const float* __restrict__ gate,
                   float* __restrict__ outL,     // d_out used as scratch
                   float* __restrict__ lend)
{
    const int blk = blockIdx.x;
    const int b   = blk / NC_;
    const int c   = blk % NC_;
    const int tid = threadIdx.x;
    const long base = (long)b * L_ + (long)c * T_;

    __shared__ float sh[2][RT_][D_];   // two 32KB tile buffers

    if (tid == 0)
        tdm_load_tile(x + base * D_, &sh[0][0][0], RT_);

    float y0 = 0.f, y1 = 0.f, y2 = 0.f, y3 = 0.f;

    for (int t = 0; t < NT_; ++t) {
        // (1) everyone finished reading buf[(t+1)&1] (iteration t-1) before
        //     the TDM is allowed to overwrite it
        __syncthreads();
        if (tid == 0) {
            if (t + 1 < NT_) {
                tdm_load_tile(x + (base + (long)(t + 1) * RT_) * D_,
                              &sh[(t + 1) & 1][0][0], RT_);
                __builtin_amdgcn_s_wait_tensorcnt(1);   // tile t landed
            } else {
                __builtin_amdgcn_s_wait_tensorcnt(0);
            }
        }
        // (2) release: tile t is in LDS
        __syncthreads();

#pragma unroll
        for (int kk = 0; kk < RT_; ++kk) {
            const long r  = base + (long)t * RT_ + kk;
            const float mu = rowmu[r];
            const float rs = rowrstd[r];
            const float a  = gate[r];
            const float om = 1.0f - a;

            const float4 v = *(const float4*)(&sh[t & 1][kk][tid * 4]);
            y0 = a * y0 + om * (v.x - mu) * rs;
            y1 = a * y1 + om * (v.y - mu) * rs;
            y2 = a * y2 + om * (v.z - mu) * rs;
            y3 = a * y3 + om * (v.w - mu) * rs;

            float4 o; o.x = y0; o.y = y1; o.z = y2; o.w = y3;
            *(float4*)(outL + r * D_ + tid * 4) = o;
        }
    }

    float4 e; e.x = y0; e.y = y1; e.z = y2; e.w = y3;
    *(float4*)(lend + (long)(b * NC_ + c) * D_ + tid * 4) = e;
}

// ---------------------------------------------------------------------------
// K2b: inter-chunk scan of carries: Yin[c] = Aend[c-1]*Yin[c-1] + Lend[c-1].
// ---------------------------------------------------------------------------
__global__ __launch_bounds__(TPB)
void k2b_interchunk(const float* __restrict__ aend,
                    const float* __restrict__ lend,
                    float* __restrict__ yin)
{
    const int g = blockIdx.x * TPB + threadIdx.x;
    if (g >= B_ * D_) return;
    const int b = g / D_;
    const int d = g % D_;
    float y = 0.0f;
    for (int c = 0; c < NC_; ++c) {
        const long idx = (long)(b * NC_ + c) * D_ + d;
        yin[idx] = y;
        y = aend[b * NC_ + c] * y + lend[idx];
    }
}

// ---------------------------------------------------------------------------
// K2c: y = A_t * Yin + local, per-row ||y|| block reduction (wave32 butterfly
//      + cross-wave LDS, parity buffered), safe-normalize, d_out in place.
// ---------------------------------------------------------------------------
__global__ __launch_bounds__(TPB)
void k2c_finalize(const float* __restrict__ avec,
                  const float* __restrict__ yin,
                  float* __restrict__ out)
{
    const int blk  = blockIdx.x;
    const int b    = blk / NC_;
    const int c    = blk % NC_;
    const int tid  = threadIdx.x;
    const int wave = tid >> 5;
    const int lane = tid & 31;
    const long base = (long)b * L_ + (long)c * T_;

    __shared__ float ssh[2][WAVES];

    const float4 w = *(const float4*)(yin + (long)(b * NC_ + c) * D_ + tid * 4);

    for (int k = 0; k < T_; ++k) {
        const long r = base + k;
        const float A = avec[r];
        float4 v = *(const float4*)(out + r * D_ + tid * 4);
        v.x = fmaf(A, w.x, v.x);
        v.y = fmaf(A, w.y, v.y);
        v.z = fmaf(A, w.z, v.z);
        v.w = fmaf(A, w.w, v.w);

        float ss = v.x*v.x + v.y*v.y + v.z*v.z + v.w*v.w;
        ss = wave_reduce_add(ss);
        if (lane == 0) ssh[k & 1][wave] = ss;
        __syncthreads();
        float S = 0.f;
#pragma unroll
        for (int wv = 0; wv < WAVES; ++wv) S += ssh[k & 1][wv];

        const float invn = 1.0f / fmaxf(sqrtf(S), NORM_EPS);
        v.x *= invn; v.y *= invn; v.z *= invn; v.w *= invn;
        *(float4*)(out + r * D_ + tid * 4) = v;
    }
}

// ---------------------------------------------------------------------------
extern "C" void kernel_launch(void* const* d_in, const int* in_sizes, int n_in,
                              void* d_out, int out_size, void* d_ws, size_t ws_size,
                              hipStream_t stream) {
    (void)in_sizes; (void)n_in; (void)out_size; (void)ws_size;

    const float* x       = (const float*)d_in[0];
    // d_in[1] = blend_logit: stats-only in the reference, unused in the output
    const float* bias_p  = (const float*)d_in[2];
    const float* scale_p = (const float*)d_in[3];
    float* out = (float*)d_out;

    // workspace layout (~4.7 MB total)
    float* ws     = (float*)d_ws;
    float* gate   = ws;                              // NROWS
    float* rowmu  = gate  + NROWS;                   // NROWS
    float* rowrs  = rowmu + NROWS;                   // NROWS
    float* avec   = rowrs + NROWS;                   // NROWS
    float* aend   = avec  + NROWS;                   // B_*NC_
    float* lend   = aend  + B_ * NC_;                // B_*NC_*D_
    float* yin    = lend  + (size_t)B_ * NC_ * D_;   // B_*NC_*D_

    const dim3 grid(B_ * NC_), tpb(TPB);
    hipLaunchKernelGGL(k1_gates,      grid, tpb, 0, stream,
                       x, bias_p, scale_p, gate, rowmu, rowrs, avec, aend);
    hipLaunchKernelGGL(k2a_localscan, grid, tpb, 0, stream,
                       x, rowmu, rowrs, gate, out, lend);
    hipLaunchKernelGGL(k2b_interchunk, dim3((B_ * D_ + TPB - 1) / TPB), tpb, 0, stream,
                       aend, lend, yin);
    hipLaunchKernelGGL(k2c_finalize,  grid, tpb, 0, stream,
                       avec, yin, out);
}